// PseudoLabel_48619029790963
// MI455X (gfx1250) — compile-verified
//
#include <hip/hip_runtime.h>
#include <cstdint>

#define CONF_THRESH 0.9f

// ---------------------------------------------------------------------------
// Async DMA: copy one 2KB row (512 f32) from global to LDS.
// 4 x global_load_async_to_lds_b128: each instr moves 32 lanes x 16B = 512B.
// Tracked on ASYNCcnt; non-temporal hint since the 256MB stream > 192MB L2.
// ---------------------------------------------------------------------------
__device__ __forceinline__ void async_copy_row_2kb(uint32_t lds_base,
                                                   const float* __restrict__ gsrc,
                                                   int lane) {
  unsigned long long ga = (unsigned long long)(uintptr_t)gsrc + (unsigned)(lane * 16);
  uint32_t la = lds_base + (uint32_t)(lane * 16);
#pragma unroll
  for (int i = 0; i < 4; ++i) {
    asm volatile("global_load_async_to_lds_b128 %0, %1, off th:TH_LOAD_NT"
                 :: "v"(la), "v"(ga)
                 : "memory");
    ga += 512ull;
    la += 512u;
  }
}

__device__ __forceinline__ float wave_max_f32(float x) {
#pragma unroll
  for (int o = 16; o > 0; o >>= 1) x = fmaxf(x, __shfl_xor(x, o, 32));
  return x;
}
__device__ __forceinline__ float wave_sum_f32(float x) {
#pragma unroll
  for (int o = 16; o > 0; o >>= 1) x += __shfl_xor(x, o, 32);
  return x;
}

// ---------------------------------------------------------------------------
// Main kernel: each block = 8 waves, each wave owns 8 rows (block = 64 rows).
// Fast path (C==512, full block): double-buffered async global->LDS pipeline.
// Per row: max, sum(exp(x-max)), p_max = 1/S; accumulate (p,1) iff p > 0.9.
// ---------------------------------------------------------------------------
__global__ __launch_bounds__(256) void pseudo_label_main(
    const float* __restrict__ pred, const float* __restrict__ mask,
    float* __restrict__ part_tot, unsigned* __restrict__ part_cnt,
    int N, int C) {
  __shared__ float rowbuf[8][2][512];   // 8 waves x double buffer x 2KB row
  __shared__ float red_t[8];
  __shared__ unsigned red_c[8];

  const int tid  = threadIdx.x;
  const int w    = tid >> 5;
  const int lane = tid & 31;
  const int blk  = blockIdx.x;
  const int row0 = blk * 64 + w * 8;

  float    wtot = 0.0f;
  unsigned wcnt = 0u;

  const bool fast = (C == 512) && (row0 + 8 <= N);
  if (fast) {
    // Low 32 bits of a generic pointer to LDS == LDS byte offset (aperture in hi bits).
    const uint32_t lb0 = (uint32_t)(uintptr_t)&rowbuf[w][0][0];
    const uint32_t lb1 = (uint32_t)(uintptr_t)&rowbuf[w][1][0];

    async_copy_row_2kb(lb0, pred + (size_t)(row0 + 0) * 512, lane);
    async_copy_row_2kb(lb1, pred + (size_t)(row0 + 1) * 512, lane);

#pragma unroll
    for (int t = 0; t < 8; ++t) {
      if (t < 7) asm volatile("s_wait_asynccnt 0x4" ::: "memory");  // row t landed
      else       asm volatile("s_wait_asynccnt 0x0" ::: "memory");  // last row

      const float m = mask[row0 + t];
      const float* rb = &rowbuf[w][t & 1][0];

      // Conflict-free LDS reads: bank(lane + 32j) = lane or lane+32, all distinct.
      float v[16];
#pragma unroll
      for (int j = 0; j < 16; ++j) v[j] = rb[lane + 32 * j] * m;

      // Drain DS reads before the DMA may overwrite this buffer.
      asm volatile("s_wait_dscnt 0x0" ::: "memory");
      if (t + 2 < 8)
        async_copy_row_2kb((t & 1) ? lb1 : lb0,
                           pred + (size_t)(row0 + t + 2) * 512, lane);

      float mx = v[0];
#pragma unroll
      for (int j = 1; j < 16; ++j) mx = fmaxf(mx, v[j]);
      mx = wave_max_f32(mx);

      float s = 0.0f;
#pragma unroll
      for (int j = 0; j < 16; ++j) s += __expf(v[j] - mx);
      s = wave_sum_f32(s);

      const float p = 1.0f / s;   // max softmax probability of this row
      if (p > CONF_THRESH) { wtot += p; wcnt += 1u; }
    }
  } else {
    // Tail / general-C fallback: two-pass nontemporal global reads.
    for (int t = 0; t < 8; ++t) {
      const int row = row0 + t;
      if (row >= N) break;
      const float m = mask[row];
      const float* rp = pred + (size_t)row * C;
      float mx = -__builtin_inff();
      for (int k = lane; k < C; k += 32)
        mx = fmaxf(mx, __builtin_nontemporal_load(rp + k) * m);
      mx = wave_max_f32(mx);
      float s = 0.0f;
      for (int k = lane; k < C; k += 32)
        s += __expf(__builtin_nontemporal_load(rp + k) * m - mx);
      s = wave_sum_f32(s);
      const float p = 1.0f / s;
      if (p > CONF_THRESH) { wtot += p; wcnt += 1u; }
    }
  }

  if (lane == 0) { red_t[w] = wtot; red_c[w] = wcnt; }
  __syncthreads();
  if (tid == 0) {
    float tt = 0.0f; unsigned cc = 0u;
#pragma unroll
    for (int i = 0; i < 8; ++i) { tt += red_t[i]; cc += red_c[i]; }
    part_tot[blk] = tt;   // every block writes its slot -> no ws memset needed
    part_cnt[blk] = cc;
  }
}

// ---------------------------------------------------------------------------
// Finalize: deterministic fixed-order reduction of per-block partials.
// ---------------------------------------------------------------------------
__global__ __launch_bounds__(256) void pseudo_label_final(
    const float* __restrict__ part_tot, const unsigned* __restrict__ part_cnt,
    float* __restrict__ out, int nparts) {
  __shared__ float st[256];
  __shared__ unsigned sc[256];
  const int tid = threadIdx.x;
  float t = 0.0f; unsigned c = 0u;
  for (int i = tid; i < nparts; i += 256) { t += part_tot[i]; c += part_cnt[i]; }
  st[tid] = t; sc[tid] = c;
  __syncthreads();
  for (int o = 128; o > 0; o >>= 1) {
    if (tid < o) { st[tid] += st[tid + o]; sc[tid] += sc[tid + o]; }
    __syncthreads();
  }
  if (tid == 0) {
    const float total = st[0];
    const unsigned cnt = sc[0];
    out[0] = (cnt > 0u) ? (-total / (float)cnt) : 0.0f;
  }
}

extern "C" void kernel_launch(void* const* d_in, const int* in_sizes, int n_in,
                              void* d_out, int out_size, void* d_ws, size_t ws_size,
                              hipStream_t stream) {
  (void)n_in; (void)out_size; (void)ws_size;
  const float* pred = (const float*)d_in[0];
  const float* mask = (const float*)d_in[1];
  const int N = in_sizes[1];           // T*B rows (mask is [T,B,1])
  const int C = in_sizes[0] / N;       // classes per row (512)
  const int nblocks = (N + 63) / 64;   // 64 rows per block

  float*    pt = (float*)d_ws;         // nblocks f32 partial totals
  unsigned* pc = (unsigned*)(pt + nblocks);

  pseudo_label_main<<<nblocks, 256, 0, stream>>>(pred, mask, pt, pc, N, C);
  pseudo_label_final<<<1, 256, 0, stream>>>(pt, pc, (float*)d_out, nblocks);
}